// HyperAggregationBatched_73942156968397
// MI455X (gfx1250) — compile-verified
//
#include <hip/hip_runtime.h>
#include <hip/hip_bf16.h>

typedef __attribute__((ext_vector_type(2))) float v2f;
typedef __attribute__((ext_vector_type(8))) float v8f;

#define H_DIM 128
#define M_DIM 8

__device__ __forceinline__ float gelu_exact(float x) {
    return 0.5f * x * (1.0f + erff(x * 0.70710678118654752440f));
}

// ---------------------------------------------------------------------------
// Kernel 1: per-node transform  T[n] = gelu(X[n] @ W_A^T) @ W_B^T   (N x 8)
// One wave (32 lanes) computes a 16-node tile with V_WMMA_F32_16X16X4_F32.
// ---------------------------------------------------------------------------
__global__ void __launch_bounds__(256) node_transform_kernel(
    const float* __restrict__ X, const float* __restrict__ W_A,
    const float* __restrict__ W_B, float* __restrict__ T, int N)
{
    __shared__ float lds[8][16 * 132];   // per-wave 16x128 tile, padded stride

    const int lane = threadIdx.x & 31;
    const int wave = threadIdx.x >> 5;
    const int l15  = lane & 15;
    const int hl   = lane >> 4;                  // which half-wave (K split)
    const int n0   = (blockIdx.x * 8 + wave) * 16;
    const bool active = (n0 < N);                // uniform per wave

    if (active) {
        // ---- GEMM1: Y(16x128) = Xtile(16x128) @ W_A^T via WMMA f32 16x16x4
        v8f acc[8] = {};                         // 8 column tiles of 16 j's
        const float* xrow = X + (size_t)(n0 + l15) * H_DIM;
        const int kb = hl * 2;                   // lanes 16-31 carry K+2,K+3

        for (int ks = 0; ks < 32; ++ks) {
            const int k = ks * 4 + kb;
            v2f a;
            a.x = xrow[k];
            a.y = xrow[k + 1];
            #pragma unroll
            for (int jt = 0; jt < 8; ++jt) {
                const int j = jt * 16 + l15;
                v2f b;                           // b[k][j] = W_A[j][k]
                b.x = W_A[j * H_DIM + k];
                b.y = W_A[j * H_DIM + k + 1];
                acc[jt] = __builtin_amdgcn_wmma_f32_16x16x4_f32(
                    false, a, false, b, (short)0, acc[jt], false, false);
            }
        }
        // ---- exact gelu + transpose-store to LDS (row-major 16x128)
        #pragma unroll
        for (int jt = 0; jt < 8; ++jt) {
            const int j = jt * 16 + l15;
            #pragma unroll
            for (int r = 0; r < 8; ++r) {
                const int m = r + hl * 8;        // D layout: lanes16-31 -> M+8
                lds[wave][m * 132 + j] = gelu_exact(acc[jt][r]);
            }
        }
    }
    __syncthreads();
    if (active) {
        // ---- GEMM2 (tiny): T(16x8) = gelu(Y) @ W_B^T via VALU
        const int nl = lane >> 1;                // node row (2 lanes per row)
        const int mb = (lane & 1) * 4;           // 4 mix outputs per lane
        const float* g  = &lds[wave][nl * 132];
        const float* w0 = W_B + (size_t)(mb + 0) * H_DIM;
        const float* w1 = W_B + (size_t)(mb + 1) * H_DIM;
        const float* w2 = W_B + (size_t)(mb + 2) * H_DIM;
        const float* w3 = W_B + (size_t)(mb + 3) * H_DIM;
        float s0 = 0.f, s1 = 0.f, s2 = 0.f, s3 = 0.f;
        for (int j = 0; j < H_DIM; ++j) {
            const float gv = g[j];
            s0 = fmaf(gv, w0[j], s0);
            s1 = fmaf(gv, w1[j], s1);
            s2 = fmaf(gv, w2[j], s2);
            s3 = fmaf(gv, w3[j], s3);
        }
        float4 r = make_float4(s0, s1, s2, s3);
        *(float4*)(T + (size_t)(n0 + nl) * M_DIM + mb) = r;
    }
}

// ---------------------------------------------------------------------------
// Kernel 2: per-edge scatter. One wave per edge.
//   agg[dst][m][h] += X[src][h] * T[src][m]   (layout (N,8,128) for coalesced
//   128B-cacheline atomics), S[dst][m] += T[src][m], count[dst] += 1
// ---------------------------------------------------------------------------
__global__ void __launch_bounds__(256) edge_scatter_kernel(
    const int* __restrict__ ei, const float* __restrict__ X,
    const float* __restrict__ T, float* __restrict__ agg,
    float* __restrict__ S, float* __restrict__ cnt, int E)
{
    const int lane = threadIdx.x & 31;
    const int wave = threadIdx.x >> 5;
    const int e = blockIdx.x * 8 + wave;
    if (e >= E) return;

    const int src = ei[e];
    const int dst = ei[E + e];

    const float4 t03 = *(const float4*)(T + (size_t)src * M_DIM);
    const float4 t47 = *(const float4*)(T + (size_t)src * M_DIM + 4);
    float t[8] = { t03.x, t03.y, t03.z, t03.w, t47.x, t47.y, t47.z, t47.w };

    const float* xp = X + (size_t)src * H_DIM + lane;
    const float x0 = xp[0], x1 = xp[32], x2 = xp[64], x3 = xp[96];

    float* ap = agg + (size_t)dst * (H_DIM * M_DIM) + lane;
    #pragma unroll
    for (int m = 0; m < M_DIM; ++m) {
        const float tm = t[m];
        atomicAdd(ap + m * H_DIM +  0, x0 * tm);
        atomicAdd(ap + m * H_DIM + 32, x1 * tm);
        atomicAdd(ap + m * H_DIM + 64, x2 * tm);
        atomicAdd(ap + m * H_DIM + 96, x3 * tm);
    }
    if (lane < M_DIM) atomicAdd(S + (size_t)dst * M_DIM + lane, t[lane]);
    if (lane == 0)    atomicAdd(cnt + dst, 1.0f);
}

// ---------------------------------------------------------------------------
// Kernel 3: out[v][h] = (sum_m gelu(agg[v][m][h]) * S[v][m]) / max(count,1)
// One wave per node.
// ---------------------------------------------------------------------------
__global__ void __launch_bounds__(256) node_output_kernel(
    const float* __restrict__ agg, const float* __restrict__ S,
    const float* __restrict__ cnt, float* __restrict__ out, int N)
{
    const int lane = threadIdx.x & 31;
    const int wave = threadIdx.x >> 5;
    const int v = blockIdx.x * 8 + wave;
    if (v >= N) return;

    float s[8];
    #pragma unroll
    for (int m = 0; m < M_DIM; ++m) s[m] = S[(size_t)v * M_DIM + m];
    const float c = fmaxf(cnt[v], 1.0f);

    const float* ap = agg + (size_t)v * (H_DIM * M_DIM) + lane;
    float* op = out + (size_t)v * H_DIM + lane;
    #pragma unroll
    for (int q = 0; q < 4; ++q) {
        float a = 0.0f;
        #pragma unroll
        for (int m = 0; m < M_DIM; ++m)
            a = fmaf(gelu_exact(ap[m * H_DIM + q * 32]), s[m], a);
        op[q * 32] = a / c;
    }
}

// ---------------------------------------------------------------------------
extern "C" void kernel_launch(void* const* d_in, const int* in_sizes, int n_in,
                              void* d_out, int out_size, void* d_ws, size_t ws_size,
                              hipStream_t stream) {
    const float* X   = (const float*)d_in[0];
    const int*   ei  = (const int*)d_in[1];   // edge_index (2,E), int32
    const float* W_A = (const float*)d_in[2];
    const float* W_B = (const float*)d_in[3];

    const int N = in_sizes[0] / H_DIM;
    const int E = in_sizes[1] / 2;

    // workspace: agg (N*8*128) | S (N*8) | count (N) | T (N*8)
    float* agg = (float*)d_ws;
    float* S   = agg + (size_t)N * (H_DIM * M_DIM);
    float* cnt = S   + (size_t)N * M_DIM;
    float* T   = cnt + (size_t)N;

    const size_t zero_bytes =
        ((size_t)N * (H_DIM * M_DIM) + (size_t)N * M_DIM + (size_t)N) * sizeof(float);
    hipMemsetAsync(d_ws, 0, zero_bytes, stream);

    node_transform_kernel<<<(N + 127) / 128, 256, 0, stream>>>(X, W_A, W_B, T, N);
    edge_scatter_kernel <<<(E + 7) / 8,     256, 0, stream>>>(ei, X, T, agg, S, cnt, E);
    node_output_kernel  <<<(N + 7) / 8,     256, 0, stream>>>(agg, S, cnt, (float*)d_out, N);
}